// GCN_37709812859638
// MI455X (gfx1250) — compile-verified
//
#include <hip/hip_runtime.h>
#include <hip/hip_bf16.h>
#include <math.h>

typedef __attribute__((ext_vector_type(2))) float v2f;
typedef __attribute__((ext_vector_type(8))) float v8f;

#define KCHUNK 32

// ---------------------------------------------------------------- utilities

__global__ void zero_f32(float* __restrict__ p, int n) {
  int i = blockIdx.x * blockDim.x + threadIdx.x;
  if (i < n) p[i] = 0.0f;
}

__global__ void deg_count(const int* __restrict__ dst, float* __restrict__ deg, int E) {
  int e = blockIdx.x * blockDim.x + threadIdx.x;
  if (e < E) unsafeAtomicAdd(&deg[dst[e]], 1.0f);
}

__global__ void deg_rsqrt(float* __restrict__ deg, int N) {
  int n = blockIdx.x * blockDim.x + threadIdx.x;
  if (n < N) deg[n] = rsqrtf(deg[n] + 1.0f);  // self-loop included in degree
}

__global__ void edge_norm(const int* __restrict__ src, const int* __restrict__ dst,
                          const float* __restrict__ dis, float* __restrict__ nrm, int E) {
  int e = blockIdx.x * blockDim.x + threadIdx.x;
  if (e < E) nrm[e] = dis[src[e]] * dis[dst[e]];
}

// ------------------------------------------------------- WMMA fp32 GEMM
// C[N,Fout] = A[N,K] @ B[K,Fout].  Block = 128 threads (4 waves), tile 64x64.
// Wave w computes rows [rowBase+16w, +16) x 64 cols (4 accumulators).
// A fragment (16x4 f32): lane l holds row l&15; VGPR pair = K = (l>>4)*2+{0,1}.
// B fragment (4x16 f32): lane l holds col l&15; VGPR pair = K = (l>>4)*2+{0,1}.
// C/D (16x16 f32): VGPR r -> M = r + 8*(lane>=16), N = lane&15.
//
// B panel is stored K-pair-interleaved in LDS:  Bs[k/2][col][2] = {B[k][c],B[k+1][c]}
// so every B fragment is ONE aligned ds_load_b64 straight into an even VGPR
// pair (no v_mov repacking in the WMMA inner loop).

__global__ __launch_bounds__(128) void gemm_wmma_f32(
    const float* __restrict__ A, const float* __restrict__ B, float* __restrict__ C,
    int N, int K, int Fout) {
  __shared__ __align__(16) float As[64 * KCHUNK];        // [64 rows][KCHUNK]
  __shared__ __align__(16) float Bs[(KCHUNK / 2) * 128]; // [KCHUNK/2][64 cols][2]

  const int tid   = threadIdx.x;
  const int wave  = tid >> 5;
  const int lane  = tid & 31;
  const int l16   = lane & 15;
  const int khalf = (lane >> 4) << 1;               // 0 or 2
  const int rowBase = blockIdx.x * 64;
  const int colBase = blockIdx.y * 64;

  v8f acc0 = {}, acc1 = {}, acc2 = {}, acc3 = {};

  for (int ko = 0; ko < K; ko += KCHUNK) {
    __syncthreads();
    // Stage A panel: 64 rows x KCHUNK  (512 float4, 4 per thread)
#pragma unroll
    for (int i = 0; i < 4; ++i) {
      int f4 = tid + i * 128;
      int r = f4 >> 3, kq = (f4 & 7) << 2;
      int rg = rowBase + r; if (rg >= N) rg = N - 1;   // clamp: garbage rows never stored
      *(float4*)(As + r * KCHUNK + kq) =
          *(const float4*)(A + (size_t)rg * K + ko + kq);
    }
    // Stage B panel K-pair interleaved: unit = (k pair, col pair) -> one float4
#pragma unroll
    for (int i = 0; i < 4; ++i) {
      int u  = tid + i * 128;        // 0..511
      int kp = u >> 5;               // k pair index 0..15
      int c  = (u & 31) << 1;        // col 0,2,...,62
      const float* gp = B + (size_t)(ko + kp * 2) * Fout + colBase + c;
      float2 g0 = *(const float2*)(gp);
      float2 g1 = *(const float2*)(gp + Fout);
      float4 wv = { g0.x, g1.x, g0.y, g1.y };
      *(float4*)(Bs + kp * 128 + c * 2) = wv;
    }
    __syncthreads();

#pragma unroll
    for (int kk = 0; kk < KCHUNK; kk += 4) {
      v2f a = *(const v2f*)(As + (wave * 16 + l16) * KCHUNK + kk + khalf);
      const v2f* bp = (const v2f*)(Bs + ((kk + khalf) >> 1) * 128 + l16 * 2);
      v2f b0 = bp[0];    // cols nt*16 + l16, K pair ready-packed
      v2f b1 = bp[16];
      v2f b2 = bp[32];
      v2f b3 = bp[48];
      acc0 = __builtin_amdgcn_wmma_f32_16x16x4_f32(false, a, false, b0, (short)0, acc0, false, false);
      acc1 = __builtin_amdgcn_wmma_f32_16x16x4_f32(false, a, false, b1, (short)0, acc1, false, false);
      acc2 = __builtin_amdgcn_wmma_f32_16x16x4_f32(false, a, false, b2, (short)0, acc2, false, false);
      acc3 = __builtin_amdgcn_wmma_f32_16x16x4_f32(false, a, false, b3, (short)0, acc3, false, false);
    }
  }

  const int rowW = rowBase + wave * 16 + ((lane >> 4) << 3);
#pragma unroll
  for (int r = 0; r < 8; ++r) {
    int row = rowW + r;
    if (row < N) {
      float* cp = C + (size_t)row * Fout + colBase + l16;
      cp[0]  = acc0[r];
      cp[16] = acc1[r];
      cp[32] = acc2[r];
      cp[48] = acc3[r];
    }
  }
}

// -------------------------------------------- edge scatter (segment_sum)
// thread-linear over E*F; a wave32 covers 32 consecutive features of one edge
// -> uniform src/dst/norm per wave, coalesced h row read, f32 HW atomics.

__global__ void scatter_edges(const int* __restrict__ src, const int* __restrict__ dst,
                              const float* __restrict__ nrm, const float* __restrict__ h,
                              float* __restrict__ agg, int E, int flog2) {
  unsigned gid = blockIdx.x * blockDim.x + threadIdx.x;
  unsigned e = gid >> flog2;
  if (e < (unsigned)E) {
    unsigned F = 1u << flog2;
    unsigned f = gid & (F - 1u);
    float v = nrm[e] * h[(size_t)src[e] * F + f];
    unsafeAtomicAdd(&agg[(size_t)dst[e] * F + f], v);
  }
}

// ------------------------- self-loop + bias + (optional) leaky-relu, in place

__global__ void combine_bias_act(const float* __restrict__ agg, const float* __restrict__ t,
                                 const float* __restrict__ dis, const float* __restrict__ bias,
                                 float* __restrict__ out, int N, int flog2, int do_lrelu) {
  unsigned gid = blockIdx.x * blockDim.x + threadIdx.x;
  unsigned n = gid >> flog2;
  if (n < (unsigned)N) {
    unsigned f = gid & ((1u << flog2) - 1u);
    float d = dis[n];
    float v = agg[gid] + d * d * t[gid] + bias[f];
    if (do_lrelu) v = (v > 0.0f) ? v : 0.01f * v;
    out[gid] = v;
  }
}

// ------------------------------------------- softmax over 64 logits per row
// one wave32 per row; 2 values per lane; shfl_xor tree reductions.

__global__ __launch_bounds__(256) void softmax64(float* __restrict__ out, int N) {
  int n = blockIdx.x * 8 + (threadIdx.x >> 5);
  int lane = threadIdx.x & 31;
  if (n >= N) return;  // uniform per wave
  float* row = out + (size_t)n * 64;
  float v0 = row[lane], v1 = row[lane + 32];
  float m = fmaxf(v0, v1);
#pragma unroll
  for (int off = 16; off > 0; off >>= 1) m = fmaxf(m, __shfl_xor(m, off, 32));
  float e0 = expf(v0 - m), e1 = expf(v1 - m);
  float s = e0 + e1;
#pragma unroll
  for (int off = 16; off > 0; off >>= 1) s += __shfl_xor(s, off, 32);
  float inv = 1.0f / s;
  row[lane]      = e0 * inv;
  row[lane + 32] = e1 * inv;
}

// ---------------------------------------------------------------- driver

extern "C" void kernel_launch(void* const* d_in, const int* in_sizes, int n_in,
                              void* d_out, int out_size, void* d_ws, size_t ws_size,
                              hipStream_t stream) {
  const float* x  = (const float*)d_in[0];
  const int*   ei = (const int*)d_in[1];
  const float* W[4]  = {(const float*)d_in[2], (const float*)d_in[4],
                        (const float*)d_in[6], (const float*)d_in[8]};
  const float* bs[4] = {(const float*)d_in[3], (const float*)d_in[5],
                        (const float*)d_in[7], (const float*)d_in[9]};

  const int FIN = 512, H = 256, CLS = 64;
  const int N = in_sizes[0] / FIN;
  const int E = in_sizes[1] / 2;
  const int* src = ei;
  const int* dst = ei + E;

  // workspace carve-out (recomputed every call; no persistent state)
  char* w = (char*)d_ws;
  size_t off = 0;
  auto take = [&](size_t bytes) -> float* {
    float* p = (float*)(w + off);
    off += (bytes + 255) & ~(size_t)255;
    return p;
  };
  float* dis  = take((size_t)N * sizeof(float));       // degree -> rsqrt(deg)
  float* nrm  = take((size_t)E * sizeof(float));       // per-edge norm
  float* bufT = take((size_t)N * H * sizeof(float));   // transform output
  float* bufA = take((size_t)N * H * sizeof(float));   // aggregation / next input

  const int TB = 256;
  zero_f32 <<<(N + TB - 1) / TB, TB, 0, stream>>>(dis, N);
  deg_count<<<(E + TB - 1) / TB, TB, 0, stream>>>(dst, dis, E);
  deg_rsqrt<<<(N + TB - 1) / TB, TB, 0, stream>>>(dis, N);
  edge_norm<<<(E + TB - 1) / TB, TB, 0, stream>>>(src, dst, dis, nrm, E);

  const int rblocks = (N + 63) / 64;
  const float* hin = x;
  int fin = FIN;
  for (int layer = 0; layer < 4; ++layer) {
    const int fout  = (layer == 3) ? CLS : H;
    const int flog2 = (layer == 3) ? 6 : 8;

    gemm_wmma_f32<<<dim3(rblocks, fout / 64), 128, 0, stream>>>(hin, W[layer], bufT, N, fin, fout);

    const unsigned nf = (unsigned)N * (unsigned)fout;
    zero_f32<<<(nf + TB - 1) / TB, TB, 0, stream>>>(bufA, (int)nf);

    const unsigned ethreads = (unsigned)E << flog2;
    scatter_edges<<<(ethreads + TB - 1) / TB, TB, 0, stream>>>(src, dst, nrm, bufT, bufA, E, flog2);

    float* outp = (layer == 3) ? (float*)d_out : bufA;
    combine_bias_act<<<(nf + TB - 1) / TB, TB, 0, stream>>>(bufA, bufT, dis, bs[layer], outp,
                                                            N, flog2, layer != 3 ? 1 : 0);
    hin = bufA;
    fin = fout;
  }

  softmax64<<<(N + 7) / 8, 256, 0, stream>>>((float*)d_out, N);
}